// cif_predictor_19275813224889
// MI455X (gfx1250) — compile-verified
//
#include <hip/hip_runtime.h>
#include <cstdint>

typedef __attribute__((ext_vector_type(2))) float v2f;
typedef __attribute__((ext_vector_type(8))) float v8f;

#define B_ 16
#define T_ 2048
#define D_ 512

// ---------------------------------------------------------------------------
// Kernel 1: fused depthwise conv(K=3) + residual + ReLU + (D->1 linear) +
// sigmoid, producing alphas[B,T].  One wave (32 lanes) handles 16 timesteps.
// The D=512 reduction runs as 128 chained V_WMMA_F32_16X16X4_F32 ops:
//   A[m=t][k=d-chunk] = relu(conv + residual),  B[k][n] = lin_w[k] (all cols)
//   -> C[m][0] = logit[t]  (exact fp32, matches reference arithmetic)
// Boundary padding is handled branch-free: neighbor rows are loaded from a
// clamped (always-valid) address and multiplied by a 0/1 lane mask, so the
// inner loop has no exec-mask branches.
// ---------------------------------------------------------------------------
__global__ __launch_bounds__(32) void alphas_kernel(
    const float* __restrict__ hidden, const float* __restrict__ conv_w,
    const float* __restrict__ conv_b, const float* __restrict__ lin_w,
    const float* __restrict__ lin_b, float* __restrict__ out_alpha)
{
  __shared__ float sW[D_ * 3];
  __shared__ float sCB[D_];
  __shared__ float sLW[D_];
  const int lane = threadIdx.x;
  for (int i = lane; i < D_ * 3; i += 32) sW[i] = conv_w[i];
  for (int i = lane; i < D_; i += 32) { sCB[i] = conv_b[i]; sLW[i] = lin_w[i]; }
  __syncthreads();

  const int b    = blockIdx.y;
  const int t0   = blockIdx.x * 16;
  const int m    = lane & 15;   // A-matrix row (time within tile)
  const int half = lane >> 4;   // K-pair selector per ISA 16x4 f32 A layout
  const int t    = t0 + m;

  // Branch-free zero padding: clamp row index, scale by 0/1 mask.
  const int   tm1   = (t > 0) ? t - 1 : 0;
  const int   tp1   = (t < T_ - 1) ? t + 1 : T_ - 1;
  const float maskL = (t > 0) ? 1.f : 0.f;
  const float maskR = (t < T_ - 1) ? 1.f : 0.f;

  const float* __restrict__ rowC = hidden + ((size_t)b * T_ + t)   * D_;
  const float* __restrict__ rowL = hidden + ((size_t)b * T_ + tm1) * D_;
  const float* __restrict__ rowR = hidden + ((size_t)b * T_ + tp1) * D_;

  v8f c = {};  // fp32 16x16 accumulator (column 0 carries the logits)
  for (int k0 = 0; k0 < D_; k0 += 4) {
    const int d = k0 + 2 * half;                  // this lane's two channels
    const float2 h0  = *(const float2*)(rowC + d);
    float2 hm1       = *(const float2*)(rowL + d);
    float2 hp1       = *(const float2*)(rowR + d);
    hm1.x *= maskL; hm1.y *= maskL;
    hp1.x *= maskR; hp1.y *= maskR;

    float o0 = sW[d * 3 + 0] * hm1.x + sW[d * 3 + 1] * h0.x +
               sW[d * 3 + 2] * hp1.x + sCB[d] + h0.x;
    float o1 = sW[(d + 1) * 3 + 0] * hm1.y + sW[(d + 1) * 3 + 1] * h0.y +
               sW[(d + 1) * 3 + 2] * hp1.y + sCB[d + 1] + h0.y;
    o0 = fmaxf(o0, 0.f);
    o1 = fmaxf(o1, 0.f);

    v2f a;  a.x = o0;        a.y = o1;           // A[m][k], k = 2*half+{0,1}
    v2f bw; bw.x = sLW[d];   bw.y = sLW[d + 1];  // B[k][n] = lin_w[k], all n
    c = __builtin_amdgcn_wmma_f32_16x16x4_f32(
        /*neg_a=*/false, a, /*neg_b=*/false, bw,
        /*c_mod=*/(short)0, c, /*reuse_a=*/false, /*reuse_b=*/false);
  }

  // Column N=0 lives in lane 0 (M=0..7) and lane 16 (M=8..15).
  if (m == 0) {
    const float lb = lin_b[0];
    const int base = half * 8;
#pragma unroll
    for (int i = 0; i < 8; ++i) {
      const float logit = c[i] + lb;
      out_alpha[(size_t)b * T_ + t0 + base + i] = 1.f / (1.f + __expf(-logit));
    }
  }
}

// ---------------------------------------------------------------------------
// Kernel 2: exact sequential integrate-and-fire scalar scan (one thread per
// batch).  Emits fires[B,T] (pre-reset integrator), token_num[B], and the
// per-step coefficient cur[B,T] + fire positions/counts into workspace.
// ---------------------------------------------------------------------------
__global__ void scan_kernel(const float* __restrict__ alphas,
                            float* __restrict__ fires,
                            float* __restrict__ token_num,
                            float* __restrict__ cur,
                            int* __restrict__ firePos,
                            int* __restrict__ nFires)
{
  const int b = threadIdx.x;
  if (b >= B_) return;
  float integ = 0.f, tok = 0.f;
  int nf = 0;
  const float* ab = alphas + (size_t)b * T_;
  for (int t = 0; t < T_; ++t) {
    const float a = ab[t];
    tok += a;
    const float dist = 1.f - integ;
    integ += a;
    fires[(size_t)b * T_ + t] = integ;          // recorded pre-reset
    const bool fire = integ >= 1.f;
    const float cu = fire ? dist : a;
    cur[(size_t)b * T_ + t] = cu;
    if (fire) {
      integ -= 1.f;                              // exact for integ in [1,2)
      firePos[(size_t)b * T_ + nf] = t;
      ++nf;
    }
  }
  token_num[b] = tok;
  nFires[b] = nf;
}

// ---------------------------------------------------------------------------
// Kernel 3: parallel ragged extraction.  Block (j, b) builds output token j:
//   sum_{t in (t_{j-1}, t_j]} cur_t * h_t  +  rem_{j-1} * h_{t_{j-1}}
// Rows j >= n_fires[b] are zero (left-packed, zero-padded gather).
// ---------------------------------------------------------------------------
__global__ __launch_bounds__(128) void extract_kernel(
    const float* __restrict__ hidden, const float* __restrict__ alphas,
    const float* __restrict__ cur, const int* __restrict__ firePos,
    const int* __restrict__ nFires, float* __restrict__ out_ac, int L)
{
  const int b = blockIdx.y;
  const int j = blockIdx.x;
  const int ch = threadIdx.x * 4;
  float4 acc = make_float4(0.f, 0.f, 0.f, 0.f);

  const int nf = nFires[b];
  if (j < nf) {
    const int tj = firePos[(size_t)b * T_ + j];
    const int tprev = (j > 0) ? firePos[(size_t)b * T_ + j - 1] : -1;
    for (int t = tprev + 1; t <= tj; ++t) {
      const float cw = cur[(size_t)b * T_ + t];
      const float4 h4 = *(const float4*)(hidden + ((size_t)b * T_ + t) * D_ + ch);
      acc.x += cw * h4.x; acc.y += cw * h4.y;
      acc.z += cw * h4.z; acc.w += cw * h4.w;
    }
    if (j > 0) {  // remainder carried over from the previous fire step
      const float rem = alphas[(size_t)b * T_ + tprev] - cur[(size_t)b * T_ + tprev];
      const float4 h4 = *(const float4*)(hidden + ((size_t)b * T_ + tprev) * D_ + ch);
      acc.x += rem * h4.x; acc.y += rem * h4.y;
      acc.z += rem * h4.z; acc.w += rem * h4.w;
    }
  }
  *(float4*)(out_ac + ((size_t)b * L + j) * D_ + ch) = acc;
}

// ---------------------------------------------------------------------------
extern "C" void kernel_launch(void* const* d_in, const int* in_sizes, int n_in,
                              void* d_out, int out_size, void* d_ws, size_t ws_size,
                              hipStream_t stream) {
  const float* hidden = (const float*)d_in[0];  // [B,T,D]
  const float* conv_w = (const float*)d_in[1];  // [D,1,3]
  const float* conv_b = (const float*)d_in[2];  // [D]
  const float* lin_w  = (const float*)d_in[3];  // [1,D]
  const float* lin_b  = (const float*)d_in[4];  // [1]

  const int B = B_, T = T_, D = D_;
  // out = concat(acoustic[B,L,D], token_num[B], alphas[B,T], fires[B,T])
  const int L = (out_size - B - 2 * B * T) / (B * D);

  float* out_ac    = (float*)d_out;
  float* out_tok   = out_ac + (size_t)B * L * D;
  float* out_alpha = out_tok + B;
  float* out_fires = out_alpha + (size_t)B * T;

  float* ws_cur     = (float*)d_ws;                    // B*T floats
  int*   ws_firePos = (int*)(ws_cur + (size_t)B * T);  // B*T ints
  int*   ws_nFires  = ws_firePos + (size_t)B * T;      // B ints

  dim3 g1(T / 16, B);
  alphas_kernel<<<g1, 32, 0, stream>>>(hidden, conv_w, conv_b, lin_w, lin_b,
                                       out_alpha);

  scan_kernel<<<1, 32, 0, stream>>>(out_alpha, out_fires, out_tok, ws_cur,
                                    ws_firePos, ws_nFires);

  if (L > 0) {
    dim3 g3(L, B);
    extract_kernel<<<g3, 128, 0, stream>>>(hidden, out_alpha, ws_cur,
                                           ws_firePos, ws_nFires, out_ac, L);
  }
}